// RelationScore_16947940950247
// MI455X (gfx1250) — compile-verified
//
#include <hip/hip_runtime.h>
#include <hip/hip_bf16.h>
#include <math.h>

typedef float v2f __attribute__((ext_vector_type(2)));
typedef float v8f __attribute__((ext_vector_type(8)));

#define TM 128         // block tile M
#define TN 64          // block tile N
#define KC 32          // K chunk staged in LDS
#define LDK 34         // padded K stride in LDS (even -> 8B aligned, conflict-free)

// ---------------------------------------------------------------------------
// Fused GEMM: C[M,N] = op(A[M,K] @ W[K,N] + bias), op = relu or identity.
// 256 threads = 8 waves; waves tiled 4(M) x 2(N); each wave computes 32x32
// via 2x2 WMMA f32 16x16x4 accumulators. Requires N % 64 == 0, K % 32 == 0.
// M is guarded (for the means GEMM where M = 64 < TM).
// ---------------------------------------------------------------------------
__global__ __launch_bounds__(256)
void gemm_wmma_f32(const float* __restrict__ A, const float* __restrict__ W,
                   const float* __restrict__ bias, float* __restrict__ C,
                   int M, int N, int K, int do_relu)
{
    __shared__ float sA[TM * LDK];   // A tile, K-contiguous per row
    __shared__ float sB[TN * LDK];   // W tile transposed: sB[n][k]

    const int tid   = threadIdx.x;
    const int lane  = tid & 31;
    const int wave  = tid >> 5;
    const int wm    = (wave & 3) * 32;   // wave offset in M within block tile
    const int wn    = (wave >> 2) * 32;  // wave offset in N within block tile
    const int l16   = lane & 15;
    const int lh    = lane >> 4;         // 0 or 1 (half-wave)

    const int blockM = blockIdx.y * TM;
    const int blockN = blockIdx.x * TN;

    v8f acc00 = {}; v8f acc01 = {}; v8f acc10 = {}; v8f acc11 = {};

    for (int k0 = 0; k0 < K; k0 += KC) {
        // ---- stage A tile: TM x KC floats = 1024 float4, 4 per thread ----
        #pragma unroll
        for (int t = 0; t < 4; ++t) {
            int idx = tid + t * 256;        // 0..1023
            int row = idx >> 3;             // KC/4 = 8 float4 per row
            int c4  = idx & 7;
            float4 v = make_float4(0.f, 0.f, 0.f, 0.f);
            int gr = blockM + row;
            if (gr < M)
                v = *(const float4*)(A + (size_t)gr * K + k0 + c4 * 4);
            float* dst = &sA[row * LDK + c4 * 4];
            dst[0] = v.x; dst[1] = v.y; dst[2] = v.z; dst[3] = v.w;
        }
        // ---- stage W tile: KC x TN -> transposed sB[n][k], 2 float4/thread ----
        #pragma unroll
        for (int t = 0; t < 2; ++t) {
            int idx = tid + t * 256;        // 0..511
            int row = idx >> 4;             // k within chunk (0..31)
            int c4  = idx & 15;             // n/4
            float4 v = *(const float4*)(W + (size_t)(k0 + row) * N + blockN + c4 * 4);
            sB[(c4 * 4 + 0) * LDK + row] = v.x;
            sB[(c4 * 4 + 1) * LDK + row] = v.y;
            sB[(c4 * 4 + 2) * LDK + row] = v.z;
            sB[(c4 * 4 + 3) * LDK + row] = v.w;
        }
        __syncthreads();

        // ---- 8 k-steps of 4 WMMAs each ----
        #pragma unroll
        for (int ks = 0; ks < KC / 4; ++ks) {
            int kk = ks * 4 + lh * 2;   // lane-half selects K pair per ISA layout
            v2f a0 = *(const v2f*)&sA[(wm + l16     ) * LDK + kk];
            v2f a1 = *(const v2f*)&sA[(wm + 16 + l16) * LDK + kk];
            v2f b0 = *(const v2f*)&sB[(wn + l16     ) * LDK + kk];
            v2f b1 = *(const v2f*)&sB[(wn + 16 + l16) * LDK + kk];
            acc00 = __builtin_amdgcn_wmma_f32_16x16x4_f32(false, a0, false, b0,
                        (short)0, acc00, false, false);
            acc01 = __builtin_amdgcn_wmma_f32_16x16x4_f32(false, a0, false, b1,
                        (short)0, acc01, false, false);
            acc10 = __builtin_amdgcn_wmma_f32_16x16x4_f32(false, a1, false, b0,
                        (short)0, acc10, false, false);
            acc11 = __builtin_amdgcn_wmma_f32_16x16x4_f32(false, a1, false, b1,
                        (short)0, acc11, false, false);
        }
        __syncthreads();
    }

    // ---- epilogue: C/D layout = VGPR r -> row r + 8*lh, col l16 ----
    auto store_tile = [&](v8f acc, int mOff, int nOff) {
        int col = blockN + nOff + l16;
        float bv = bias ? bias[col] : 0.f;
        #pragma unroll
        for (int r = 0; r < 8; ++r) {
            int row = blockM + mOff + r + 8 * lh;
            if (row < M) {
                float x = acc[r] + bv;
                if (do_relu) x = fmaxf(x, 0.f);
                C[(size_t)row * N + col] = x;
            }
        }
    };
    store_tile(acc00, wm,      wn);
    store_tile(acc01, wm,      wn + 16);
    store_tile(acc10, wm + 16, wn);
    store_tile(acc11, wm + 16, wn + 16);
}

// ---------------------------------------------------------------------------
// Deterministic class means: one block per class (64), 256 threads cover the
// 512 embedding dims (2 each). Scalar label loads, no atomics.
// ---------------------------------------------------------------------------
__global__ __launch_bounds__(256)
void class_means_kernel(const float* __restrict__ s_emb,
                        const int* __restrict__ labels,
                        float* __restrict__ means)
{
    const int c   = blockIdx.x;
    const int tid = threadIdx.x;
    float acc0 = 0.f, acc1 = 0.f;
    int count = 0;
    for (int i = 0; i < 4096; ++i) {
        if (labels[i] == c) {
            const float* row = s_emb + (size_t)i * 512;
            acc0 += row[tid];
            acc1 += row[tid + 256];
            ++count;
        }
    }
    float inv = 1.0f / fmaxf((float)count, 1.0f);
    means[(size_t)c * 512 + tid]       = acc0 * inv;
    means[(size_t)c * 512 + tid + 256] = acc1 * inv;
}

// ---------------------------------------------------------------------------
// Final fused stage: out[q,c] = sigmoid(b2 + sum_k relu(Aq[q,k]+Bc[c,k])*W2[k])
// Block handles 16 queries x 64 classes; Aq tile + W2 in LDS; Bc (64KB total)
// streams through cache. Each thread owns one class and 4 query rows.
// ---------------------------------------------------------------------------
#define QT 16
__global__ __launch_bounds__(256)
void final_scores_kernel(const float* __restrict__ Aq, const float* __restrict__ Bc,
                         const float* __restrict__ W2, const float* __restrict__ b2,
                         float* __restrict__ out)
{
    __shared__ float sA[QT * 256];  // 16 KB
    __shared__ float sW[256];

    const int tid   = threadIdx.x;
    const int qBase = blockIdx.x * QT;

    #pragma unroll
    for (int t = 0; t < QT; ++t)
        sA[t * 256 + tid] = Aq[(size_t)(qBase + t) * 256 + tid];
    sW[tid] = W2[tid];
    __syncthreads();

    const int c  = tid & 63;
    const int q0 = tid >> 6;                    // 0..3 -> rows q0, q0+4, q0+8, q0+12
    const float* brow = Bc + (size_t)c * 256;

    float s0 = 0.f, s1 = 0.f, s2 = 0.f, s3 = 0.f;
    for (int k = 0; k < 256; ++k) {
        float b = brow[k];
        float w = sW[k];
        s0 += fmaxf(sA[(q0     ) * 256 + k] + b, 0.f) * w;
        s1 += fmaxf(sA[(q0 +  4) * 256 + k] + b, 0.f) * w;
        s2 += fmaxf(sA[(q0 +  8) * 256 + k] + b, 0.f) * w;
        s3 += fmaxf(sA[(q0 + 12) * 256 + k] + b, 0.f) * w;
    }
    float bb = b2[0];
    out[(size_t)(qBase + q0     ) * 64 + c] = 1.f / (1.f + __expf(-(s0 + bb)));
    out[(size_t)(qBase + q0 +  4) * 64 + c] = 1.f / (1.f + __expf(-(s1 + bb)));
    out[(size_t)(qBase + q0 +  8) * 64 + c] = 1.f / (1.f + __expf(-(s2 + bb)));
    out[(size_t)(qBase + q0 + 12) * 64 + c] = 1.f / (1.f + __expf(-(s3 + bb)));
}

// ---------------------------------------------------------------------------
extern "C" void kernel_launch(void* const* d_in, const int* in_sizes, int n_in,
                              void* d_out, int out_size, void* d_ws, size_t ws_size,
                              hipStream_t stream)
{
    (void)in_sizes; (void)n_in; (void)out_size; (void)ws_size;

    const float* support = (const float*)d_in[0];   // (4096, 1024)
    const float* query   = (const float*)d_in[1];   // (8192, 1024)
    const int*   labels  = (const int*)  d_in[2];   // (4096,)
    const float* W_e     = (const float*)d_in[3];   // (1024, 512)
    const float* b_e     = (const float*)d_in[4];   // (512,)
    const float* W1      = (const float*)d_in[5];   // (1024, 256)
    const float* b1      = (const float*)d_in[6];   // (256,)
    const float* W2      = (const float*)d_in[7];   // (256, 1)
    const float* b2      = (const float*)d_in[8];   // (1,)
    float* out = (float*)d_out;                     // (8192, 64)

    const float* Wq = W1;              // rows 0..511 of W1
    const float* Wc = W1 + 512 * 256;  // rows 512..1023 of W1

    // workspace layout (floats)
    float* ws     = (float*)d_ws;
    float* s_emb  = ws;                       // 4096*512  = 2,097,152
    float* q_emb  = s_emb + 4096 * 512;       // 8192*512  = 4,194,304
    float* means  = q_emb + 8192 * 512;       // 64*512    =    32,768
    float* Aq     = means + 64 * 512;         // 8192*256  = 2,097,152
    float* Bc     = Aq    + 8192 * 256;       // 64*256    =    16,384
                                              // total ~33.7 MB

    dim3 blk(256);

    // 1) embeddings: relu(X @ W_e + b_e)
    gemm_wmma_f32<<<dim3(512 / TN, 4096 / TM), blk, 0, stream>>>(
        support, W_e, b_e, s_emb, 4096, 512, 1024, 1);
    gemm_wmma_f32<<<dim3(512 / TN, 8192 / TM), blk, 0, stream>>>(
        query, W_e, b_e, q_emb, 8192, 512, 1024, 1);

    // 2) class means (deterministic scan, no atomics)
    class_means_kernel<<<dim3(64), blk, 0, stream>>>(s_emb, labels, means);

    // 3) Aq = q_emb @ Wq  (no bias, no relu)
    gemm_wmma_f32<<<dim3(256 / TN, 8192 / TM), blk, 0, stream>>>(
        q_emb, Wq, nullptr, Aq, 8192, 256, 512, 0);

    // 4) Bc = means @ Wc + b1  (b1 folded here; M=64 guarded inside)
    gemm_wmma_f32<<<dim3(256 / TN, 1), blk, 0, stream>>>(
        means, Wc, b1, Bc, 64, 256, 512, 0);

    // 5) out = sigmoid(b2 + sum_k relu(Aq + Bc) * W2)
    final_scores_kernel<<<dim3(8192 / QT), blk, 0, stream>>>(Aq, Bc, W2, b2, out);
}